// CacheGNN_70970039599202
// MI455X (gfx1250) — compile-verified
//
#include <hip/hip_runtime.h>

typedef __attribute__((ext_vector_type(2))) float v2f;
typedef __attribute__((ext_vector_type(8))) float v8f;

#define N_NODES 50000
#define N_EDGES 800000
#define FEAT 128
#define HID 64
#define N_CLS 12

// ---------------------------------------------------------------------------
// utility kernels
// ---------------------------------------------------------------------------
__global__ void zero_f32(float* __restrict__ p, int n) {
    int i = blockIdx.x * blockDim.x + threadIdx.x;
    if (i < n) p[i] = 0.0f;
}

__global__ void degree_kernel(const int* __restrict__ dst, float* __restrict__ deg,
                              int n_edges) {
    int e = blockIdx.x * blockDim.x + threadIdx.x;
    if (e < n_edges) atomicAdd(&deg[dst[e]], 1.0f);
}

// in-place: dinv = rsqrt(deg + 1)
__global__ void dinv_kernel(float* __restrict__ deg_dinv, int n) {
    int i = blockIdx.x * blockDim.x + threadIdx.x;
    if (i < n) deg_dinv[i] = rsqrtf(deg_dinv[i] + 1.0f);
}

// ---------------------------------------------------------------------------
// WMMA GEMM: Y[rows x (NT*16)] = X[rows x K] @ W[K x (NT*16)]
// One wave computes a 16-row strip; W staged in LDS; f32 16x16x4 WMMA.
// A layout : lane(0..15)=M, V0=K0,V1=K1 ; lanes 16..31 -> K2,K3
// B layout : lane&15=N,    V0=K0,V1=K1 ; lanes 16..31 -> K2,K3
// C/D      : 8 VGPRs, M = r (+8 for lanes>=16), N = lane&15
// ---------------------------------------------------------------------------
template <int K, int NT>
__global__ __launch_bounds__(256) void wmma_gemm_kernel(
    const float* __restrict__ X, const float* __restrict__ W,
    float* __restrict__ Y, int n_row_tiles) {
    constexpr int NCOLS = NT * 16;
    __shared__ float wlds[K * NCOLS];
    for (int i = threadIdx.x; i < K * NCOLS; i += 256) wlds[i] = W[i];
    __syncthreads();

    const int wave = threadIdx.x >> 5;
    const int lane = threadIdx.x & 31;
    const int tile = blockIdx.x * 8 + wave;
    if (tile >= n_row_tiles) return;  // uniform per-wave: EXEC stays all-ones

    const int m    = lane & 15;         // row within A-tile / col within B-tile
    const int koff = (lane >> 4) << 1;  // 0 for lanes 0-15, 2 for lanes 16-31
    const float* xrow = X + (size_t)(tile * 16 + m) * K;

    v8f acc[NT];
#pragma unroll
    for (int t = 0; t < NT; ++t) { v8f z = {}; acc[t] = z; }

#pragma unroll 4
    for (int k = 0; k < K; k += 4) {
        v2f a;
        a.x = xrow[k + koff];
        a.y = xrow[k + koff + 1];
#pragma unroll
        for (int t = 0; t < NT; ++t) {
            v2f b;
            b.x = wlds[(k + koff) * NCOLS + t * 16 + m];
            b.y = wlds[(k + koff + 1) * NCOLS + t * 16 + m];
            acc[t] = __builtin_amdgcn_wmma_f32_16x16x4_f32(
                false, a, false, b, (short)0, acc[t], false, false);
        }
    }

    const int rbase = tile * 16 + ((lane >> 4) << 3);  // +8 for upper half-wave
#pragma unroll
    for (int t = 0; t < NT; ++t)
#pragma unroll
        for (int r = 0; r < 8; ++r)
            Y[(size_t)(rbase + r) * NCOLS + t * 16 + m] = acc[t][r];
}

// ---------------------------------------------------------------------------
// Edge aggregation: agg[dst] += xw[src] * dinv[src]*dinv[dst]
// one thread per (edge, feature); 64 consecutive threads read one source row
// ---------------------------------------------------------------------------
__global__ __launch_bounds__(256) void edge_agg_kernel(
    const int* __restrict__ src, const int* __restrict__ dst,
    const float* __restrict__ dinv, const float* __restrict__ xw,
    float* __restrict__ agg, long long n_work) {
    long long idx = (long long)blockIdx.x * blockDim.x + threadIdx.x;
    if (idx >= n_work) return;
    int e = (int)(idx >> 6);
    int f = (int)(idx & 63);
    int s = src[e];
    int d = dst[e];
    float norm = dinv[s] * dinv[d];
    atomicAdd(&agg[(size_t)d * HID + f], xw[(size_t)s * HID + f] * norm);
}

// ---------------------------------------------------------------------------
// h = relu(agg + xw * dinv^2 + b)
// ---------------------------------------------------------------------------
__global__ __launch_bounds__(256) void self_relu_kernel(
    const float* __restrict__ agg, const float* __restrict__ xw,
    const float* __restrict__ dinv, const float* __restrict__ bias,
    float* __restrict__ h, int n_elems) {
    int i = blockIdx.x * blockDim.x + threadIdx.x;
    if (i >= n_elems) return;
    int n = i >> 6;
    int f = i & 63;
    float di = dinv[n];
    float v = agg[i] + xw[i] * (di * di) + bias[f];
    h[i] = v > 0.0f ? v : 0.0f;
}

// ---------------------------------------------------------------------------
// pad Wfc [64x12] -> [64x16] (zeros in cols 12..15)
// ---------------------------------------------------------------------------
__global__ void pad_wfc_kernel(const float* __restrict__ Wfc, float* __restrict__ wpad) {
    int i = blockIdx.x * blockDim.x + threadIdx.x;
    if (i >= HID * 16) return;
    int k = i >> 4;
    int n = i & 15;
    wpad[i] = (n < N_CLS) ? Wfc[k * N_CLS + n] : 0.0f;
}

// ---------------------------------------------------------------------------
// final FC: out[rows x 12] = H[rows x 64] @ Wpad[64 x 16] + bfc (store 12 cols)
// ---------------------------------------------------------------------------
__global__ __launch_bounds__(256) void wmma_fc_kernel(
    const float* __restrict__ H, const float* __restrict__ Wpad,
    const float* __restrict__ bfc, float* __restrict__ out, int n_row_tiles) {
    __shared__ float wlds[HID * 16];
    for (int i = threadIdx.x; i < HID * 16; i += 256) wlds[i] = Wpad[i];
    __syncthreads();

    const int wave = threadIdx.x >> 5;
    const int lane = threadIdx.x & 31;
    const int tile = blockIdx.x * 8 + wave;
    if (tile >= n_row_tiles) return;

    const int m    = lane & 15;
    const int koff = (lane >> 4) << 1;
    const float* hrow = H + (size_t)(tile * 16 + m) * HID;

    v8f acc = {};
#pragma unroll
    for (int k = 0; k < HID; k += 4) {
        v2f a;
        a.x = hrow[k + koff];
        a.y = hrow[k + koff + 1];
        v2f b;
        b.x = wlds[(k + koff) * 16 + m];
        b.y = wlds[(k + koff + 1) * 16 + m];
        acc = __builtin_amdgcn_wmma_f32_16x16x4_f32(
            false, a, false, b, (short)0, acc, false, false);
    }

    if (m < N_CLS) {  // divergence only after all WMMAs are done
        float bb = bfc[m];
        const int rbase = tile * 16 + ((lane >> 4) << 3);
#pragma unroll
        for (int r = 0; r < 8; ++r)
            out[(size_t)(rbase + r) * N_CLS + m] = acc[r] + bb;
    }
}

// ---------------------------------------------------------------------------
// driver
// ---------------------------------------------------------------------------
extern "C" void kernel_launch(void* const* d_in, const int* in_sizes, int n_in,
                              void* d_out, int out_size, void* d_ws, size_t ws_size,
                              hipStream_t stream) {
    const float* x   = (const float*)d_in[0];
    const int*   ei  = (const int*)d_in[1];     // [2, N_EDGES], row-major
    const float* W1  = (const float*)d_in[2];
    const float* b1  = (const float*)d_in[3];
    const float* W2  = (const float*)d_in[4];
    const float* b2  = (const float*)d_in[5];
    const float* Wfc = (const float*)d_in[6];
    const float* bfc = (const float*)d_in[7];
    float* out = (float*)d_out;

    const int* src = ei;
    const int* dst = ei + N_EDGES;

    // workspace layout (floats): A | B | C | dinv | wpad
    float* A    = (float*)d_ws;            // xw buffer      [N_NODES*HID]
    float* B    = A + (size_t)N_NODES * HID;  // agg buffer  [N_NODES*HID]
    float* C    = B + (size_t)N_NODES * HID;  // h buffer    [N_NODES*HID]
    float* dinv = C + (size_t)N_NODES * HID;  // deg -> dinv [N_NODES]
    float* wpad = dinv + N_NODES;             // [HID*16]

    const int nf_elems   = N_NODES * HID;                 // 3.2M
    const int nf_blocks  = (nf_elems + 255) / 256;
    const int node_blk   = (N_NODES + 255) / 256;
    const int edge_blk   = (N_EDGES + 255) / 256;
    const long long ework = (long long)N_EDGES * HID;     // 51.2M
    const int ework_blk  = (int)((ework + 255) / 256);
    const int row_tiles  = N_NODES / 16;                  // 3125 exact
    const int gemm_blk   = (row_tiles + 7) / 8;

    // ---- degrees & symmetric norm ----
    zero_f32<<<node_blk, 256, 0, stream>>>(dinv, N_NODES);
    degree_kernel<<<edge_blk, 256, 0, stream>>>(dst, dinv, N_EDGES);
    dinv_kernel<<<node_blk, 256, 0, stream>>>(dinv, N_NODES);

    // ---- layer 1 ----
    wmma_gemm_kernel<FEAT, HID / 16><<<gemm_blk, 256, 0, stream>>>(x, W1, A, row_tiles);
    zero_f32<<<nf_blocks, 256, 0, stream>>>(B, nf_elems);
    edge_agg_kernel<<<ework_blk, 256, 0, stream>>>(src, dst, dinv, A, B, ework);
    self_relu_kernel<<<nf_blocks, 256, 0, stream>>>(B, A, dinv, b1, C, nf_elems);

    // ---- layer 2 ----
    wmma_gemm_kernel<HID, HID / 16><<<gemm_blk, 256, 0, stream>>>(C, W2, A, row_tiles);
    zero_f32<<<nf_blocks, 256, 0, stream>>>(B, nf_elems);
    edge_agg_kernel<<<ework_blk, 256, 0, stream>>>(src, dst, dinv, A, B, ework);
    self_relu_kernel<<<nf_blocks, 256, 0, stream>>>(B, A, dinv, b2, C, nf_elems);

    // ---- classifier ----
    pad_wfc_kernel<<<(HID * 16 + 255) / 256, 256, 0, stream>>>(Wfc, wpad);
    wmma_fc_kernel<<<gemm_blk, 256, 0, stream>>>(C, wpad, bfc, out, row_tiles);
}